// TemporalHeteroGNN_74225624809923
// MI455X (gfx1250) — compile-verified
//
#include <hip/hip_runtime.h>
#include <math.h>

#define NUSER  30000
#define NEVENT 60000
#define NVENUE 1000
#define NTOTAL (NUSER + NEVENT + NVENUE)

typedef __attribute__((ext_vector_type(2))) float v2f;
typedef __attribute__((ext_vector_type(8))) float v8f;

// ---------------------------------------------------------------------------
// fp32 WMMA GEMM, K and N compile-time:
//   C[m][n] = sum_k A[m][k]*B[k][n] + bias[n] (+ emb[ids[m]][n])
// One wave per 16x16 tile; fully unrolled chain of V_WMMA_F32_16X16X4_F32.
// OOB M rows handled by clamped-row loads * 0/1 mask (no EXEC divergence).
// ---------------------------------------------------------------------------
template <int K, int N>
__global__ void wmma_gemm_kernel(const float* __restrict__ A, int lda,
                                 const float* __restrict__ B,   // K x N row-major
                                 const float* __restrict__ bias,
                                 float* __restrict__ C, int ldc,
                                 int M,
                                 const float* __restrict__ emb, // optional, ld 64
                                 const int* __restrict__ ids)
{
    const int wave   = (int)((blockIdx.x * blockDim.x + threadIdx.x) >> 5);
    const int lane   = threadIdx.x & 31;
    const int ntiles = N >> 4;
    const int mt = wave / ntiles;
    const int nt = wave - mt * ntiles;
    const int m0 = mt << 4;
    if (m0 >= M) return;

    const int half = lane >> 4;        // 0: lanes 0-15, 1: lanes 16-31
    const int l15  = lane & 15;
    const int n    = (nt << 4) + l15;

    v8f c;
    const float bv = bias ? bias[n] : 0.0f;
#pragma unroll
    for (int r = 0; r < 8; ++r) c[r] = bv;

    const int   am    = m0 + l15;
    const int   amc   = (am < M) ? am : (M - 1);         // clamp, keep loads in-bounds
    const float amask = (am < M) ? 1.0f : 0.0f;          // zero the fragment instead
    const float* __restrict__ arow = A + (size_t)amc * lda + half * 2;
    const float* __restrict__ bp   = B + (size_t)(half * 2) * N + n;

#pragma unroll
    for (int k0 = 0; k0 < K; k0 += 4) {
        v2f a = *(const v2f*)(arow + k0);                // b64 load, 8B aligned
        a[0] *= amask;
        a[1] *= amask;
        v2f b;
        b[0] = bp[k0 * N];                               // immediate offsets
        b[1] = bp[k0 * N + N];
        c = __builtin_amdgcn_wmma_f32_16x16x4_f32(false, a, false, b, (short)0, c,
                                                  false, false);
    }

#pragma unroll
    for (int r = 0; r < 8; ++r) {       // D: lane<16 -> M=r, lane>=16 -> M=r+8
        const int m = m0 + r + half * 8;
        if (m < M) {
            float v = c[r];
            if (emb) v += emb[(size_t)ids[m] * 64 + n];
            C[(size_t)m * ldc + n] = v;
        }
    }
}

// ---------------------------------------------------------------------------
// Output projection + gated skip + ReLU (K = N = 64):
//   h = relu( g * (A@Wout + bout) + (1-g) * h ),  g = sigmoid(skip)
// A is gelu(agg), applied by a prior elementwise pass.
// ---------------------------------------------------------------------------
__global__ void wmma_out_gemm_kernel(const float* __restrict__ A,
                                     const float* __restrict__ B,    // 64x64
                                     const float* __restrict__ bias, // 64
                                     const float* __restrict__ skip, // scalar
                                     float* __restrict__ H,          // ld 64, in/out
                                     int M)
{
    const int wave = (int)((blockIdx.x * blockDim.x + threadIdx.x) >> 5);
    const int lane = threadIdx.x & 31;
    const int mt = wave >> 2;            // ntiles = 64/16 = 4
    const int nt = wave & 3;
    const int m0 = mt << 4;
    if (m0 >= M) return;

    const int half = lane >> 4, l15 = lane & 15;
    const int n = (nt << 4) + l15;

    v8f c;
    const float bv = bias[n];
#pragma unroll
    for (int r = 0; r < 8; ++r) c[r] = bv;

    const int   am    = m0 + l15;
    const int   amc   = (am < M) ? am : (M - 1);
    const float amask = (am < M) ? 1.0f : 0.0f;
    const float* __restrict__ arow = A + (size_t)amc * 64 + half * 2;
    const float* __restrict__ bp   = B + (half * 2) * 64 + n;

#pragma unroll
    for (int k0 = 0; k0 < 64; k0 += 4) {
        v2f a = *(const v2f*)(arow + k0);
        a[0] *= amask;
        a[1] *= amask;
        v2f b;
        b[0] = bp[k0 * 64];
        b[1] = bp[k0 * 64 + 64];
        c = __builtin_amdgcn_wmma_f32_16x16x4_f32(false, a, false, b, (short)0, c,
                                                  false, false);
    }

    const float g = 1.0f / (1.0f + expf(-skip[0]));
#pragma unroll
    for (int r = 0; r < 8; ++r) {
        const int m = m0 + r + half * 8;
        if (m < M) {
            const float hn = g * c[r] + (1.0f - g) * H[(size_t)m * 64 + n];
            H[(size_t)m * 64 + n] = fmaxf(hn, 0.0f);
        }
    }
}

// ---------------------------------------------------------------------------
// Edge passes
// ---------------------------------------------------------------------------
__device__ inline void atomicMaxF(float* addr, float val) {
    if (val >= 0.0f) atomicMax((int*)addr, __float_as_int(val));
    else             atomicMin((unsigned int*)addr, __float_as_uint(val));
}

__device__ inline void load16(const float* __restrict__ p, float* r) {
    const float4* p4 = (const float4*)p;
    float4 a = p4[0], b = p4[1], c = p4[2], d = p4[3];
    r[0]=a.x;  r[1]=a.y;  r[2]=a.z;  r[3]=a.w;
    r[4]=b.x;  r[5]=b.y;  r[6]=b.z;  r[7]=b.w;
    r[8]=c.x;  r[9]=c.y;  r[10]=c.z; r[11]=c.w;
    r[12]=d.x; r[13]=d.y; r[14]=d.z; r[15]=d.w;
}

// Pass 1: score[e,h] = scale * p_rel[h] * sum_{d,j} k[src,h,d] Wk[h,d,j] q[dst,h,j]
__global__ void edge_score_kernel(const float* __restrict__ kqv,
                                  const int* __restrict__ esrc,
                                  const int* __restrict__ edst,
                                  const float* __restrict__ Wk,   // 4*16*16
                                  const float* __restrict__ prel, // 4
                                  int nE, int soff, int doff,
                                  float* __restrict__ sc,         // nE*4
                                  float* __restrict__ smax)       // NTOTAL*4
{
    __shared__ float sWk[1024];
    __shared__ float sPr[4];
    for (int i = threadIdx.x; i < 1024; i += blockDim.x) sWk[i] = Wk[i];
    if (threadIdx.x < 4) sPr[threadIdx.x] = prel[threadIdx.x];
    __syncthreads();

    const int tid = blockIdx.x * blockDim.x + threadIdx.x;
    const int e = tid >> 2, h = tid & 3;
    if (e >= nE) return;
    const int s = esrc[e], d = edst[e];

    float k[16], q[16];
    load16(kqv + (size_t)(soff + s) * 192 + h * 16, k);
    load16(kqv + (size_t)(doff + d) * 192 + 64 + h * 16, q);

    float acc = 0.0f;
#pragma unroll
    for (int dd = 0; dd < 16; ++dd) {
        float t = 0.0f;
#pragma unroll
        for (int j = 0; j < 16; ++j) t += sWk[(h * 16 + dd) * 16 + j] * q[j];
        acc += k[dd] * t;
    }
    const float score = acc * 0.25f * sPr[h];   // scale = 1/sqrt(16)
    sc[(size_t)e * 4 + h] = score;
    atomicMaxF(&smax[(size_t)(doff + d) * 4 + h], score);
}

// Pass 2: ex = exp(score - smax[dst,h]); den[dst,h] += ex; sc <- ex
__global__ void edge_exp_kernel(const int* __restrict__ edst, int nE, int doff,
                                float* __restrict__ sc,
                                const float* __restrict__ smax,
                                float* __restrict__ den)
{
    const int tid = blockIdx.x * blockDim.x + threadIdx.x;
    const int e = tid >> 2, h = tid & 3;
    if (e >= nE) return;
    const int d = edst[e];
    const float m  = smax[(size_t)(doff + d) * 4 + h];
    const float ex = expf(sc[(size_t)e * 4 + h] - m);
    sc[(size_t)e * 4 + h] = ex;
    atomicAdd(&den[(size_t)(doff + d) * 4 + h], ex);
}

// Pass 3: agg[dst,h,:] += (ex/den) * (v[src,h,:] @ Wv[h])
__global__ void edge_agg_kernel(const float* __restrict__ kqv,
                                const int* __restrict__ esrc,
                                const int* __restrict__ edst,
                                const float* __restrict__ Wv,  // 4*16*16
                                int nE, int soff, int doff,
                                const float* __restrict__ sc,
                                const float* __restrict__ den,
                                float* __restrict__ agg)       // NTOTAL*64
{
    __shared__ float sWv[1024];
    for (int i = threadIdx.x; i < 1024; i += blockDim.x) sWv[i] = Wv[i];
    __syncthreads();

    const int tid = blockIdx.x * blockDim.x + threadIdx.x;
    const int e = tid >> 2, h = tid & 3;
    if (e >= nE) return;
    const int s = esrc[e], d = edst[e];

    float v[16];
    load16(kqv + (size_t)(soff + s) * 192 + 128 + h * 16, v);
    const float alpha = sc[(size_t)e * 4 + h] / den[(size_t)(doff + d) * 4 + h];

    float* ap = agg + (size_t)(doff + d) * 64 + h * 16;
#pragma unroll
    for (int j = 0; j < 16; ++j) {
        float ve = 0.0f;
#pragma unroll
        for (int dd = 0; dd < 16; ++dd) ve += v[dd] * sWv[(h * 16 + dd) * 16 + j];
        atomicAdd(&ap[j], alpha * ve);
    }
}

// ---------------------------------------------------------------------------
// Elementwise helpers
// ---------------------------------------------------------------------------
__global__ void fill_kernel(float* p, float val, size_t n) {
    const size_t i = (size_t)blockIdx.x * blockDim.x + threadIdx.x;
    if (i < n) p[i] = val;
}

__global__ void gelu_kernel(float* x, size_t n) {
    const size_t i = (size_t)blockIdx.x * blockDim.x + threadIdx.x;
    if (i < n) {
        const float v = x[i];
        x[i] = 0.5f * v * (1.0f + erff(v * 0.70710678118654752f));  // exact gelu
    }
}

// wave-per-node L2 normalization, rows of 64
__global__ void l2norm_kernel(const float* __restrict__ h, float* __restrict__ out,
                              int nNodes)
{
    const int wave = (int)((blockIdx.x * blockDim.x + threadIdx.x) >> 5);
    const int lane = threadIdx.x & 31;
    if (wave >= nNodes) return;
    const float* row = h + (size_t)wave * 64;
    const float a = row[lane], b = row[lane + 32];
    float s = a * a + b * b;
#pragma unroll
    for (int off = 16; off > 0; off >>= 1) s += __shfl_xor(s, off, 32);
    const float inv = 1.0f / fmaxf(sqrtf(s), 1e-12f);
    out[(size_t)wave * 64 + lane]      = a * inv;
    out[(size_t)wave * 64 + lane + 32] = b * inv;
}

// ---------------------------------------------------------------------------
// Host
// ---------------------------------------------------------------------------
static inline int gemmBlocks(int M, int N) {
    const int waves = ((M + 15) / 16) * (N / 16);
    return (waves + 7) / 8;   // 256 threads = 8 waves per block
}

extern "C" void kernel_launch(void* const* d_in, const int* in_sizes, int n_in,
                              void* d_out, int out_size, void* d_ws, size_t ws_size,
                              hipStream_t stream)
{
    const float* x_user  = (const float*)d_in[0];
    const float* x_event = (const float*)d_in[1];
    const float* x_venue = (const float*)d_in[2];
    const float* W_user  = (const float*)d_in[3];
    const float* b_user  = (const float*)d_in[4];
    const float* W_event = (const float*)d_in[5];
    const float* b_event = (const float*)d_in[6];
    const float* W_venue = (const float*)d_in[7];
    const float* b_venue = (const float*)d_in[8];
    const float* emb_user  = (const float*)d_in[9];
    const float* emb_event = (const float*)d_in[10];
    const float* emb_venue = (const float*)d_in[11];
    const float* Wkqv  = (const float*)d_in[12];  // (2,3,64,192)
    const float* bkqv  = (const float*)d_in[13];  // (2,3,192)
    const float* Wk    = (const float*)d_in[14];  // (2,13,4,16,16)
    const float* Wv    = (const float*)d_in[15];  // (2,13,4,16,16)
    const float* p_rel = (const float*)d_in[16];  // (2,13,4)
    const float* Wout  = (const float*)d_in[17];  // (2,3,64,64)
    const float* bout  = (const float*)d_in[18];  // (2,3,64)
    const float* skip  = (const float*)d_in[19];  // (2,3)
    const int* ids_user  = (const int*)d_in[20];
    const int* ids_event = (const int*)d_in[21];
    const int* ids_venue = (const int*)d_in[22];

    static const int OFFS[4]   = {0, NUSER, NUSER + NEVENT, NTOTAL};
    static const int NT[3]     = {NUSER, NEVENT, NVENUE};
    static const int E_SRC[13] = {0,0,0,0,0,0,1,1,1,1,1,1,2};
    static const int E_DST[13] = {0,1,1,1,1,1,0,0,0,0,0,2,1};

    size_t Etot = 0;
    int nEdges[13];
    for (int e = 0; e < 13; ++e) { nEdges[e] = in_sizes[23 + e] / 2; Etot += (size_t)nEdges[e]; }

    // workspace layout (floats)
    float* ws   = (float*)d_ws;
    float* h    = ws;                                // NTOTAL*64
    float* kqv  = h    + (size_t)NTOTAL * 64;        // NTOTAL*192
    float* agg  = kqv  + (size_t)NTOTAL * 192;       // NTOTAL*64
    float* sc   = agg  + (size_t)NTOTAL * 64;        // Etot*4
    float* smax = sc   + Etot * 4;                   // NTOTAL*4
    float* den  = smax + (size_t)NTOTAL * 4;         // NTOTAL*4

    // ---- input projections: h = x @ W + b + emb[ids] -------------------
    wmma_gemm_kernel<32, 64><<<gemmBlocks(NUSER, 64), 256, 0, stream>>>(
        x_user, 32, W_user, b_user, h + (size_t)OFFS[0] * 64, 64,
        NUSER, emb_user, ids_user);
    wmma_gemm_kernel<32, 64><<<gemmBlocks(NEVENT, 64), 256, 0, stream>>>(
        x_event, 32, W_event, b_event, h + (size_t)OFFS[1] * 64, 64,
        NEVENT, emb_event, ids_event);
    wmma_gemm_kernel<16, 64><<<gemmBlocks(NVENUE, 64), 256, 0, stream>>>(
        x_venue, 16, W_venue, b_venue, h + (size_t)OFFS[2] * 64, 64,
        NVENUE, emb_venue, ids_venue);

    for (int l = 0; l < 2; ++l) {
        hipMemsetAsync(agg, 0, (size_t)NTOTAL * 64 * sizeof(float), stream);
        hipMemsetAsync(den, 0, (size_t)NTOTAL * 4 * sizeof(float), stream);
        {
            const size_t n = (size_t)NTOTAL * 4;
            fill_kernel<<<(int)((n + 255) / 256), 256, 0, stream>>>(smax, -INFINITY, n);
        }

        // kqv = h @ Wkqv[l][t] + bkqv[l][t]
        for (int t = 0; t < 3; ++t) {
            wmma_gemm_kernel<64, 192><<<gemmBlocks(NT[t], 192), 256, 0, stream>>>(
                h + (size_t)OFFS[t] * 64, 64,
                Wkqv + (size_t)(l * 3 + t) * 64 * 192,
                bkqv + (size_t)(l * 3 + t) * 192,
                kqv + (size_t)OFFS[t] * 192, 192,
                NT[t], nullptr, nullptr);
        }

        // pass 1: scores + segment max
        size_t eoff = 0;
        for (int e = 0; e < 13; ++e) {
            const int n = nEdges[e];
            const int* src = (const int*)d_in[23 + e];
            const int* dst = src + n;
            edge_score_kernel<<<(n * 4 + 255) / 256, 256, 0, stream>>>(
                kqv, src, dst,
                Wk + (size_t)(l * 13 + e) * 1024,
                p_rel + (size_t)(l * 13 + e) * 4,
                n, OFFS[E_SRC[e]], OFFS[E_DST[e]],
                sc + eoff * 4, smax);
            eoff += (size_t)n;
        }
        // pass 2: exp + segment sum
        eoff = 0;
        for (int e = 0; e < 13; ++e) {
            const int n = nEdges[e];
            const int* src = (const int*)d_in[23 + e];
            const int* dst = src + n;
            edge_exp_kernel<<<(n * 4 + 255) / 256, 256, 0, stream>>>(
                dst, n, OFFS[E_DST[e]], sc + eoff * 4, smax, den);
            eoff += (size_t)n;
        }
        // pass 3: weighted message scatter-add
        eoff = 0;
        for (int e = 0; e < 13; ++e) {
            const int n = nEdges[e];
            const int* src = (const int*)d_in[23 + e];
            const int* dst = src + n;
            edge_agg_kernel<<<(n * 4 + 255) / 256, 256, 0, stream>>>(
                kqv, src, dst,
                Wv + (size_t)(l * 13 + e) * 1024,
                n, OFFS[E_SRC[e]], OFFS[E_DST[e]],
                sc + eoff * 4, den, agg);
            eoff += (size_t)n;
        }

        // gelu(agg) elementwise (exact erf form)
        {
            const size_t n = (size_t)NTOTAL * 64;
            gelu_kernel<<<(int)((n + 255) / 256), 256, 0, stream>>>(agg, n);
        }

        // out projection + gated skip + relu, per node type
        for (int t = 0; t < 3; ++t) {
            wmma_out_gemm_kernel<<<gemmBlocks(NT[t], 64), 256, 0, stream>>>(
                agg + (size_t)OFFS[t] * 64,
                Wout + (size_t)(l * 3 + t) * 4096,
                bout + (size_t)(l * 3 + t) * 64,
                skip + (size_t)(l * 3 + t),
                h + (size_t)OFFS[t] * 64, NT[t]);
        }
    }

    // final l2 normalization -> d_out (user, event, venue concatenated = h order)
    {
        const int waves = NTOTAL;
        const int threads = waves * 32;
        l2norm_kernel<<<(threads + 255) / 256, 256, 0, stream>>>(h, (float*)d_out, NTOTAL);
    }
}